// Identity_71468255805561
// MI455X (gfx1250) — compile-verified
//
#include <hip/hip_runtime.h>
#include <hip/hip_bf16.h>
#include <stdint.h>

// ---------------------------------------------------------------------------
// one_hot(input, 16384)^2  (== one_hot)  +  pred = input
//
// Output layout (flat float32):
//   [0, 4096*16384)              : p2, row-major [256,16,16384]
//   [4096*16384, 4096*16384+4096): pred (input cast to float)
//
// Strategy: bandwidth-bound memset of ~268MB of zeros using the CDNA5 async
// LDS->global store path (ASYNCcnt-tracked GLOBAL_STORE_ASYNC_FROM_LDS_B128),
// then a tiny scatter kernel that plants the 4096 ones and the pred tail.
// ---------------------------------------------------------------------------

#define ZB_THREADS 256
#define ZB_BLOCKS  2048

__global__ __launch_bounds__(ZB_THREADS)
void zero_fill_async_kernel(float* __restrict__ out, unsigned long long n_chunks /*16B units*/)
{
    // 4KB LDS staging tile of zeros: 16 bytes per lane, reused by every
    // async store this block issues.
    __shared__ __align__(16) float zeros[ZB_THREADS * 4];

    const unsigned tid = threadIdx.x;
    *reinterpret_cast<float4*>(&zeros[tid * 4]) = make_float4(0.f, 0.f, 0.f, 0.f);
    __syncthreads();  // ds stores complete (DScnt) before the async engine reads LDS

    // Generic pointers to LDS carry the LDS byte offset in their low 32 bits
    // (flat->LDS mapping truncates the upper half), so this is the per-lane
    // LDS source address for the async store.
    const uint32_t lds_off = (uint32_t)(uintptr_t)(&zeros[tid * 4]);

    const unsigned long long base   = (unsigned long long)(uintptr_t)out;
    const unsigned long long stride = (unsigned long long)gridDim.x * ZB_THREADS;

    for (unsigned long long c = (unsigned long long)blockIdx.x * ZB_THREADS + tid;
         c < n_chunks; c += stride) {
        unsigned long long gaddr = base + (c << 4);  // lane-contiguous -> 512B/wave store
        // GLOBAL_STORE_ASYNC_FROM_LDS_B128: 16B per lane, LDS -> global,
        // fire-and-forget, tracked by ASYNCcnt.
        asm volatile("global_store_async_from_lds_b128 %0, %1, off"
                     :
                     : "v"(gaddr), "v"(lds_off)
                     : "memory");
    }

    // Drain all outstanding async stores before the wave retires.
    asm volatile("s_wait_asynccnt 0" ::: "memory");
}

__global__ void onehot_scatter_kernel(const int* __restrict__ idx,
                                      float* __restrict__ p2,
                                      float* __restrict__ pred,
                                      int n, int dict)
{
    int t = blockIdx.x * blockDim.x + threadIdx.x;
    if (t < n) {
        int k = idx[t];                       // in [0, dict)
        p2[(size_t)t * (size_t)dict + (size_t)k] = 1.0f;  // one_hot^2 == one_hot
        pred[t] = (float)k;                   // pred = input, flattened into float out
    }
}

extern "C" void kernel_launch(void* const* d_in, const int* in_sizes, int n_in,
                              void* d_out, int out_size, void* d_ws, size_t ws_size,
                              hipStream_t stream)
{
    (void)n_in; (void)d_ws; (void)ws_size; (void)out_size;

    const int* input = (const int*)d_in[0];   // [256,16] int32 indices
    // d_in[1] (teacher_forcing) unused by the reference outputs.

    const int n    = in_sizes[0];             // 4096
    const int DICT = 16384;

    float* out = (float*)d_out;
    const size_t p2_elems = (size_t)n * (size_t)DICT;        // 67,108,864 floats
    const unsigned long long n_chunks =
        (unsigned long long)(p2_elems / 4);                  // 16B chunks (exact)

    // 1) Bulk zero of p2 via async LDS->global DMA stores.
    zero_fill_async_kernel<<<ZB_BLOCKS, ZB_THREADS, 0, stream>>>(out, n_chunks);

    // 2) Plant the 4096 ones and write the pred tail (stream-ordered after #1).
    onehot_scatter_kernel<<<(n + 255) / 256, 256, 0, stream>>>(
        input, out, out + p2_elems, n, DICT);
}